// HyperbolicTransformerLayer_19516331393624
// MI455X (gfx1250) — compile-verified
//
#include <hip/hip_runtime.h>
#include <math.h>

#define NN   50000
#define EE   800000
#define DD   128
#define HH   4
#define HDIM 32
#define KD   129   // D+1 (GEMM reduction length)
#define QROW 33    // HD+1 per-head row
#define QSTR 132   // H * QROW

typedef __attribute__((ext_vector_type(2))) float v2f;
typedef __attribute__((ext_vector_type(8))) float v8f;

// ---------------------------------------------------------------------------
// Dense "space" projection: out(node, col) = sum_d X[node,d] * W[col,d] + b[col]
// X:[NN,129], W:[128,129] row-major, 128 output cols.
// One block = 16-node tile, 8 waves x 16 cols, K-loop of 33 x V_WMMA_F32_16X16X4_F32.
// Output index: node*outRS + 1 + col + (col>>5)*headPad  (headPad=1 leaves the
// per-head time slot at offset 0 of each 33-float head row; headPad=0 -> [N,129]).
// ---------------------------------------------------------------------------
__global__ __launch_bounds__(256)
void gemm_space_kernel(const float* __restrict__ X,
                       const float* __restrict__ W,
                       const float* __restrict__ bias,
                       float* __restrict__ out,
                       int outRS, int headPad)
{
    __shared__ float xs[16][132];          // 16 nodes x K (zero-padded 129->132)
    const int tid   = threadIdx.x;
    const int node0 = blockIdx.x * 16;

    for (int idx = tid; idx < 16 * 132; idx += 256) {
        int r = idx / 132, c = idx % 132;
        xs[r][c] = (c < KD) ? X[(size_t)(node0 + r) * KD + c] : 0.0f;
    }
    __syncthreads();

    const int lane  = tid & 31;
    const int wave  = tid >> 5;
    const int mrow  = lane & 15;           // A-matrix row / B,N column within tile
    const int khalf = (lane >> 4) * 2;     // K sub-offset per lane half
    const int col   = wave * 16 + mrow;    // output feature column 0..127

    v8f acc = {};
    for (int kk = 0; kk < 33; ++kk) {
        const int kA = kk * 4 + khalf;     // even by construction
        v2f a, b;
        a[0] = xs[mrow][kA];
        a[1] = xs[mrow][kA + 1];
        // clamp the address (stay in-bounds), zero the value past K=129
        float w0 = W[(size_t)col * KD + ((kA     < KD) ? kA     : (KD - 1))];
        float w1 = W[(size_t)col * KD + ((kA + 1 < KD) ? kA + 1 : (KD - 1))];
        b[0] = (kA     < KD) ? w0 : 0.0f;
        b[1] = (kA + 1 < KD) ? w1 : 0.0f;
        acc = __builtin_amdgcn_wmma_f32_16x16x4_f32(
                  false, a, false, b, (short)0, acc, false, false);
    }

    const float bcol = bias[col];
    union { v8f v; float f[8]; } u; u.v = acc;
    const int nodehi = (lane >> 4) * 8;    // C/D layout: lanes16-31 hold M=r+8
    const int cidx   = 1 + col + ((col >> 5) * headPad);
    #pragma unroll
    for (int r = 0; r < 8; ++r) {
        int node = node0 + nodehi + r;
        out[(size_t)node * outRS + cidx] = u.f[r] + bcol;
    }
}

// time coordinate per (node, head): row[0] = sqrt(1 + |space|^2)
__global__ void head_time_kernel(float* __restrict__ p, int rows)
{
    int i = blockIdx.x * blockDim.x + threadIdx.x;
    if (i >= rows) return;
    float* row = p + (size_t)i * QROW;
    float s = 0.0f;
    #pragma unroll
    for (int j = 1; j < QROW; ++j) s = fmaf(row[j], row[j], s);
    row[0] = sqrtf(s + 1.0f);
}

// per (edge, head): e = exp(score); accumulate segment sums by dst
__global__ void edge_exp_kernel(const int* __restrict__ ei,
                                const float* __restrict__ q,
                                const float* __restrict__ k,
                                float* __restrict__ ev,
                                float* __restrict__ esum)
{
    int t = blockIdx.x * blockDim.x + threadIdx.x;
    if (t >= EE * HH) return;
    int e = t >> 2, h = t & 3;
    int src = ei[e], dst = ei[EE + e];
    const float* qr = q + ((size_t)dst * HH + h) * QROW;
    const float* kr = k + ((size_t)src * HH + h) * QROW;
    float dot = 0.0f;
    #pragma unroll
    for (int j = 0; j < QROW; ++j) dot = fmaf(qr[j], kr[j], dot);
    float cin = dot - 2.0f * qr[0] * kr[0];          // Lorentz inner product
    float sc  = (2.0f + 2.0f * cin) * 0.17677669529663687f;  // / sqrt(32)
    float x   = expf(sc);                             // shift-invariant softmax
    ev[t] = x;
    atomicAdd(&esum[(size_t)dst * HH + h], x);
}

// per (edge, head): attn = e/sum; agg[dst] += attn * v[src]
__global__ void edge_agg_kernel(const int* __restrict__ ei,
                                const float* __restrict__ v,
                                const float* __restrict__ ev,
                                const float* __restrict__ esum,
                                float* __restrict__ agg)
{
    int t = blockIdx.x * blockDim.x + threadIdx.x;
    if (t >= EE * HH) return;
    int e = t >> 2, h = t & 3;
    int src = ei[e], dst = ei[EE + e];
    float a = ev[t] / esum[(size_t)dst * HH + h];
    const float* vr = v   + ((size_t)src * HH + h) * QROW;
    float*       ar = agg + ((size_t)dst * HH + h) * QROW;
    #pragma unroll
    for (int j = 0; j < QROW; ++j) atomicAdd(&ar[j], a * vr[j]);
}

// Lorentz midpoint normalization, merge heads, recompute time -> y[N,129]
__global__ void finalize_kernel(const float* __restrict__ agg,
                                float* __restrict__ y)
{
    int n = blockIdx.x * blockDim.x + threadIdx.x;
    if (n >= NN) return;
    float tot = 0.0f;
    float* yr = y + (size_t)n * KD;
    for (int h = 0; h < HH; ++h) {
        const float* ar = agg + ((size_t)n * HH + h) * QROW;
        float ss = 0.0f;
        #pragma unroll
        for (int j = 0; j < QROW; ++j) ss = fmaf(ar[j], ar[j], ss);
        float a0  = ar[0];
        float cin = fabsf(ss - 2.0f * a0 * a0);       // |cinner(agg,agg)|
        cin = fmaxf(cin, 1e-8f);
        float scale = 1.0f / sqrtf(cin);              // sqrt(K)=1
        #pragma unroll
        for (int j = 1; j < QROW; ++j) {
            float o = ar[j] * scale;
            yr[1 + h * HDIM + (j - 1)] = o;
            tot = fmaf(o, o, tot);
        }
    }
    yr[0] = sqrtf(tot + 1.0f);
}

// final time coordinate over the full 128-dim space row
__global__ void out_time_kernel(float* __restrict__ o)
{
    int n = blockIdx.x * blockDim.x + threadIdx.x;
    if (n >= NN) return;
    float* r = o + (size_t)n * KD;
    float s = 0.0f;
    #pragma unroll 8
    for (int j = 1; j < KD; ++j) s = fmaf(r[j], r[j], s);
    r[0] = sqrtf(s + 1.0f);
}

extern "C" void kernel_launch(void* const* d_in, const int* in_sizes, int n_in,
                              void* d_out, int out_size, void* d_ws, size_t ws_size,
                              hipStream_t stream)
{
    const float* x  = (const float*)d_in[0];
    const int*   ei = (const int*)  d_in[1];
    const float* Wq = (const float*)d_in[2];
    const float* bq = (const float*)d_in[3];
    const float* Wk = (const float*)d_in[4];
    const float* bk = (const float*)d_in[5];
    const float* Wv = (const float*)d_in[6];
    const float* bv = (const float*)d_in[7];
    const float* Wo = (const float*)d_in[8];
    const float* bo = (const float*)d_in[9];
    float* out = (float*)d_out;

    float* ws = (float*)d_ws;
    size_t off = 0;
    float* q    = ws + off; off += (size_t)NN * QSTR;   // [N,H,33]
    float* k    = ws + off; off += (size_t)NN * QSTR;
    float* v    = ws + off; off += (size_t)NN * QSTR;
    float* ev   = ws + off; off += (size_t)EE * HH;     // exp(scores) [E,H]
    float* esum = ws + off; off += (size_t)NN * HH;     // softmax denominators
    float* agg  = ws + off; off += (size_t)NN * QSTR;   // weighted midpoint
    float* y    = ws + off; off += (size_t)NN * KD;     // merged-head hyperboloid pt

    // zero only the accumulated regions (esum + agg are adjacent)
    hipMemsetAsync(esum, 0,
                   ((size_t)NN * HH + (size_t)NN * QSTR) * sizeof(float), stream);

    const int ntiles = NN / 16;                         // 3125, exact
    gemm_space_kernel<<<ntiles, 256, 0, stream>>>(x, Wq, bq, q, QSTR, 1);
    gemm_space_kernel<<<ntiles, 256, 0, stream>>>(x, Wk, bk, k, QSTR, 1);
    gemm_space_kernel<<<ntiles, 256, 0, stream>>>(x, Wv, bv, v, QSTR, 1);

    const int rows = NN * HH;
    const int tb   = (rows + 255) / 256;
    head_time_kernel<<<tb, 256, 0, stream>>>(q, rows);
    head_time_kernel<<<tb, 256, 0, stream>>>(k, rows);
    head_time_kernel<<<tb, 256, 0, stream>>>(v, rows);

    const int et = EE * HH;
    const int eb = (et + 255) / 256;
    edge_exp_kernel<<<eb, 256, 0, stream>>>(ei, q, k, ev, esum);
    edge_agg_kernel<<<eb, 256, 0, stream>>>(ei, v, ev, esum, agg);

    finalize_kernel<<<(NN + 255) / 256, 256, 0, stream>>>(agg, y);

    gemm_space_kernel<<<ntiles, 256, 0, stream>>>(y, Wo, bo, out, KD, 0);
    out_time_kernel<<<(NN + 255) / 256, 256, 0, stream>>>(out);
}